// AttentionAggregator_60120952209411
// MI455X (gfx1250) — compile-verified
//
#include <hip/hip_runtime.h>
#include <math.h>

typedef __attribute__((ext_vector_type(2))) float v2f;
typedef __attribute__((ext_vector_type(8))) float v8f;

#define A_DIM 128
#define B_DIM 128

// ---------------------------------------------------------------------------
// Zero the aggregation buffer (d_out) and row_sum each call (harness poisons
// once with 0xAA and never re-poisons; we must start from zeros every call).
// ---------------------------------------------------------------------------
__global__ void zero_kernel(float* __restrict__ out, float* __restrict__ row_sum,
                            int n_out, int n_rows) {
    int i = blockIdx.x * blockDim.x + threadIdx.x;
    if (i < n_out)  out[i]     = 0.0f;
    if (i < n_rows) row_sum[i] = 0.0f;
}

// ---------------------------------------------------------------------------
// new_emb = feature_b @ W + b   via V_WMMA_F32_16X16X4_F32 (exact f32 math).
// Block = 256 threads = 8 wave32s. Block computes rows [row0, row0+16) x all
// 128 columns; wave w owns the 16x16 tile at columns [16w, 16w+16).
// W (64KB) is staged in LDS once per block.
//
// f32 WMMA layouts (CDNA5 ISA 7.12.2):
//   A 16x4:  lane = M + 16*(K/2), vgpr = K%2
//   B 4x16:  lane = N + 16*(K/2), vgpr = K%2
//   C 16x16: lane = N + 16*(M/8), vgpr = M%8
// ---------------------------------------------------------------------------
__global__ __launch_bounds__(256) void gemm_wmma_kernel(
    const float* __restrict__ fb, const float* __restrict__ W,
    const float* __restrict__ bias, float* __restrict__ new_emb, int NB) {
    __shared__ float sW[B_DIM * B_DIM];
    for (int i = threadIdx.x; i < B_DIM * B_DIM; i += 256) sW[i] = W[i];
    __syncthreads();

    const int wave = threadIdx.x >> 5;
    const int lane = threadIdx.x & 31;
    const int half = lane >> 4;   // selects K pair (A) / K pair (B) / M half (C)
    const int lm   = lane & 15;   // M for A loads, N for B loads and C tile
    const int row0 = blockIdx.x * 16;
    const int col0 = wave * 16;

    // Clamp OOB rows for loads (WMMA needs EXEC all-ones); guard on store only.
    const int arow = (row0 + lm < NB) ? (row0 + lm) : (NB - 1);
    const float2* __restrict__ arowp = (const float2*)(fb + (size_t)arow * B_DIM);

    v8f c = {};
#pragma unroll
    for (int k = 0; k < B_DIM; k += 4) {
        const int ka = k + 2 * half;        // this lane's K pair
        float2 av = arowp[ka >> 1];         // A[m][ka], A[m][ka+1]
        v2f a, b;
        a[0] = av.x;
        a[1] = av.y;
        b[0] = sW[ka * B_DIM + col0 + lm];        // W[ka][n]
        b[1] = sW[(ka + 1) * B_DIM + col0 + lm];  // W[ka+1][n]
        // 8 args: (neg_a, A, neg_b, B, c_mod, C, reuse_a, reuse_b)
        c = __builtin_amdgcn_wmma_f32_16x16x4_f32(false, a, false, b,
                                                  (short)0, c, false, false);
    }

    const float bv = bias[col0 + lm];
#pragma unroll
    for (int r = 0; r < 8; ++r) {
        const int row = row0 + r + 8 * half;   // M = r + 8*half
        if (row < NB)
            new_emb[(size_t)row * B_DIM + col0 + lm] = c[r] + bv;
    }
}

// ---------------------------------------------------------------------------
// alpha_a[i] = feature_a[i] . a_vec[0:128]
// alpha_b[i] = new_emb[i]   . a_vec[128:256]
// ---------------------------------------------------------------------------
__global__ void alpha_kernel(const float* __restrict__ fa,
                             const float* __restrict__ new_emb,
                             const float* __restrict__ a_vec,
                             float* __restrict__ alpha_a,
                             float* __restrict__ alpha_b, int NA, int NB) {
    int i = blockIdx.x * blockDim.x + threadIdx.x;
    if (i < NA) {
        const float4* r = (const float4*)(fa + (size_t)i * A_DIM);
        const float4* v = (const float4*)a_vec;
        float s = 0.0f;
        for (int j = 0; j < A_DIM / 4; ++j) {
            float4 x = r[j], y = v[j];
            s += x.x * y.x + x.y * y.y + x.z * y.z + x.w * y.w;
        }
        alpha_a[i] = s;
    }
    if (i < NB) {
        const float4* r = (const float4*)(new_emb + (size_t)i * B_DIM);
        const float4* v = (const float4*)(a_vec + A_DIM);
        float s = 0.0f;
        for (int j = 0; j < B_DIM / 4; ++j) {
            float4 x = r[j], y = v[j];
            s += x.x * y.x + x.y * y.y + x.z * y.z + x.w * y.w;
        }
        alpha_b[i] = s;
    }
}

// ---------------------------------------------------------------------------
// One wave32 per edge: compute attention weight, scatter w*new_emb[dst] into
// agg[src] with f32 atomics (L2 atomic units combine; new_emb/agg fit in the
// 192MB L2, so this is the L2-resident hot loop).
// ---------------------------------------------------------------------------
__global__ __launch_bounds__(256) void edge_kernel(
    const int* __restrict__ edges, const float* __restrict__ alpha_a,
    const float* __restrict__ alpha_b, const float* __restrict__ new_emb,
    float* __restrict__ agg, float* __restrict__ row_sum, int E) {
    const int e    = (blockIdx.x * 256 + threadIdx.x) >> 5;
    const int lane = threadIdx.x & 31;
    if (e >= E) return;

    const int src = edges[2 * e];
    const int dst = edges[2 * e + 1];

    const float s = alpha_a[src] + alpha_b[dst];
    // e = exp( s>0 ? s : 0.1*expm1(min(s,0)) );  min(s,0)==s on the else path
    const float w = (s > 0.0f) ? expf(s) : expf(0.1f * expm1f(s));

    if (lane == 0) atomicAdd(row_sum + src, w);

    const float4 v = ((const float4*)(new_emb + (size_t)dst * B_DIM))[lane];
    float* o = agg + (size_t)src * B_DIM + lane * 4;
    atomicAdd(o + 0, w * v.x);
    atomicAdd(o + 1, w * v.y);
    atomicAdd(o + 2, w * v.z);
    atomicAdd(o + 3, w * v.w);
}

// ---------------------------------------------------------------------------
// out[i][:] = agg[i][:] / (row_sum[i] == 0 ? 1 : row_sum[i])   (in place)
// ---------------------------------------------------------------------------
__global__ void finalize_kernel(float* __restrict__ out,
                                const float* __restrict__ row_sum, int n) {
    int i = blockIdx.x * blockDim.x + threadIdx.x;
    if (i < n) {
        float rs = row_sum[i / B_DIM];
        out[i] = out[i] / (rs == 0.0f ? 1.0f : rs);
    }
}

extern "C" void kernel_launch(void* const* d_in, const int* in_sizes, int n_in,
                              void* d_out, int out_size, void* d_ws, size_t ws_size,
                              hipStream_t stream) {
    const int*   edges = (const int*)d_in[0];    // (E,2) int
    const float* fa    = (const float*)d_in[1];  // (NA,128) f32
    const float* fb    = (const float*)d_in[2];  // (NB,128) f32
    const float* W     = (const float*)d_in[3];  // (128,128) f32
    const float* bias  = (const float*)d_in[4];  // (128,) f32
    const float* a_vec = (const float*)d_in[5];  // (256,1) f32

    const int E  = in_sizes[0] / 2;
    const int NA = in_sizes[1] / A_DIM;
    const int NB = in_sizes[2] / B_DIM;

    float* out = (float*)d_out;                  // agg accumulator + result

    // Workspace layout: new_emb | alpha_a | alpha_b | row_sum
    float* ws       = (float*)d_ws;
    float* new_emb  = ws;                                // NB*128
    float* alpha_a  = new_emb + (size_t)NB * B_DIM;      // NA
    float* alpha_b  = alpha_a + NA;                      // NB
    float* row_sum  = alpha_b + NB;                      // NA

    const int n_out = NA * B_DIM;
    zero_kernel<<<(n_out + 255) / 256, 256, 0, stream>>>(out, row_sum, n_out, NA);

    gemm_wmma_kernel<<<(NB + 15) / 16, 256, 0, stream>>>(fb, W, bias, new_emb, NB);

    const int nmax = (NA > NB) ? NA : NB;
    alpha_kernel<<<(nmax + 255) / 256, 256, 0, stream>>>(fa, new_emb, a_vec,
                                                         alpha_a, alpha_b, NA, NB);

    // one wave32 per edge
    const long long threads_e = (long long)E * 32;
    edge_kernel<<<(int)((threads_e + 255) / 256), 256, 0, stream>>>(
        edges, alpha_a, alpha_b, new_emb, out, row_sum, E);

    finalize_kernel<<<(n_out + 255) / 256, 256, 0, stream>>>(out, row_sum, n_out);
}